// KFLoss_17051020165142
// MI455X (gfx1250) — compile-verified
//
#include <hip/hip_runtime.h>
#include <hip/hip_bf16.h>

#define BETA      (1.0f / 9.0f)
#define HALF_BETA (0.5f / 9.0f)
#define INV_2BETA 4.5f

typedef float v2f __attribute__((ext_vector_type(2)));
typedef float v8f __attribute__((ext_vector_type(8)));

__device__ __forceinline__ float smooth_l1(float d) {
    // d >= 0 here
    return (d < BETA) ? (INV_2BETA * d * d) : (d - HALF_BETA);
}

__device__ __forceinline__ void sigma_of(float w, float h, float r,
                                         float& o00, float& o01, float& o11) {
    w = fminf(fmaxf(w, 1e-7f), 1e7f);
    h = fminf(fmaxf(h, 1e-7f), 1e7f);
    float c = __cosf(r);   // v_cos_f32 path; args are in [0,1) -> accurate
    float s = __sinf(r);
    float a = 0.25f * w * w;
    float b = 0.25f * h * h;
    o00 = a * c * c + b * s * s;
    o01 = (a - b) * s * c;
    o11 = a * s * s + b * c * c;
}

__device__ __forceinline__ float kf_row(float xp, float yp, float wp, float hp, float rp,
                                        float xt, float yt, float wt, float ht, float rt) {
    float p00, p01, p11, t00, t01, t11;
    sigma_of(wp, hp, rp, p00, p01, p11);
    sigma_of(wt, ht, rt, t00, t01, t11);

    float xy_loss = smooth_l1(fabsf(xp - xt)) + smooth_l1(fabsf(yp - yt));

    float Vbp = 4.0f * sqrtf(p00 * p11 - p01 * p01);
    float Vbt = 4.0f * sqrtf(t00 * t11 - t01 * t01);

    float m00 = p00 + t00, m01 = p01 + t01, m11 = p11 + t11;
    float inv = 1.0f / (m00 * m11 - m01 * m01);
    float a00 = (p00 * m11 - p01 * m01) * inv;
    float a01 = (p01 * m00 - p00 * m01) * inv;
    float a10 = (p01 * m11 - p11 * m01) * inv;
    float a11 = (p11 * m00 - p01 * m01) * inv;

    float s00 = p00 - (a00 * p00 + a01 * p01);
    float s01 = p01 - (a00 * p01 + a01 * p11);
    float s10 = p01 - (a10 * p00 + a11 * p01);
    float s11 = p11 - (a10 * p01 + a11 * p11);

    float Vb = 4.0f * sqrtf(s00 * s11 - s01 * s10);
    if (isnan(Vb)) Vb = 0.0f;

    float kfiou = Vb / (Vbp + Vbt - Vb + 1e-6f);
    return fmaxf(xy_loss + 1.0f - kfiou, 0.0f);
}

// ---------------- Stage 1: elementwise loss + per-block partial sums ----------------
// 4 rows per thread: 20 floats = 5 aligned float4 loads per input (GLOBAL_LOAD_B128).
// Default RT temporal hints on purpose: the whole 160MB working set fits in the
// 192MB L2, so repeated graph replays should hit L2, not HBM.
__global__ __launch_bounds__(256) void kf_stage1(const float* __restrict__ pred,
                                                 const float* __restrict__ tgt,
                                                 float* __restrict__ partial,
                                                 long long N) {
    __shared__ float wsum[8];
    const long long tid  = (long long)blockIdx.x * blockDim.x + threadIdx.x;
    const long long row0 = tid * 4;

    float acc = 0.0f;
    if (row0 + 4 <= N) {
        const float4* p4 = (const float4*)pred + tid * 5;
        const float4* t4 = (const float4*)tgt  + tid * 5;
        float pf[20], tf[20];
#pragma unroll
        for (int k = 0; k < 5; ++k) {
            float4 v = p4[k];
            pf[4 * k + 0] = v.x; pf[4 * k + 1] = v.y;
            pf[4 * k + 2] = v.z; pf[4 * k + 3] = v.w;
        }
#pragma unroll
        for (int k = 0; k < 5; ++k) {
            float4 v = t4[k];
            tf[4 * k + 0] = v.x; tf[4 * k + 1] = v.y;
            tf[4 * k + 2] = v.z; tf[4 * k + 3] = v.w;
        }
#pragma unroll
        for (int r = 0; r < 4; ++r) {
            acc += kf_row(pf[5 * r + 0], pf[5 * r + 1], pf[5 * r + 2], pf[5 * r + 3], pf[5 * r + 4],
                          tf[5 * r + 0], tf[5 * r + 1], tf[5 * r + 2], tf[5 * r + 3], tf[5 * r + 4]);
        }
    } else {
        for (int r = 0; r < 4; ++r) {
            long long row = row0 + r;
            if (row < N) {
                const float* p = pred + row * 5;
                const float* t = tgt  + row * 5;
                acc += kf_row(p[0], p[1], p[2], p[3], p[4],
                              t[0], t[1], t[2], t[3], t[4]);
            }
        }
    }

    // wave32 reduction
#pragma unroll
    for (int off = 16; off > 0; off >>= 1)
        acc += __shfl_down(acc, off, 32);

    const int lane = threadIdx.x & 31;
    const int wid  = threadIdx.x >> 5;
    if (lane == 0) wsum[wid] = acc;
    __syncthreads();
    if (threadIdx.x == 0) {
        float b = 0.0f;
#pragma unroll
        for (int i = 0; i < 8; ++i) b += wsum[i];
        partial[blockIdx.x] = b;
    }
}

// ---------------- Stage 2: deterministic final reduction with WMMA ----------------
// Each lane accumulates strided partials in double, then the wave's 32 values are
// reduced by a single V_WMMA_F32_16X16X4_F32 against an all-ones B matrix:
// A layout (ISA 7.12.2): lane L<16 holds (M=L, K=0|1), lane L>=16 holds (M=L-16, K=2|3).
// With a.y = 0 and B = ones, D[m][*] = val(lane m) + val(lane m+16); summing the 8 D
// VGPRs and combining halves via shfl_xor(16) yields the full wave sum.
// EXEC is all 1s here (no divergence before the WMMA), as the ISA requires.
__global__ __launch_bounds__(256) void kf_stage2(const float* __restrict__ partial,
                                                 float* __restrict__ out,
                                                 int P, long long N) {
    __shared__ float wsum[8];

    double dacc = 0.0;
    for (int i = threadIdx.x; i < P; i += 256)
        dacc += (double)partial[i];
    float v = (float)dacc;

    v2f a; a.x = v;    a.y = 0.0f;
    v2f b; b.x = 1.0f; b.y = 1.0f;
    v8f c = {};
    c = __builtin_amdgcn_wmma_f32_16x16x4_f32(
            /*neg_a=*/false, a, /*neg_b=*/false, b,
            /*c_mod=*/(short)0, c, /*reuse_a=*/false, /*reuse_b=*/false);

    float t = c[0] + c[1] + c[2] + c[3] + c[4] + c[5] + c[6] + c[7];
    t += __shfl_xor(t, 16, 32);   // combine rows 0-7 half with rows 8-15 half

    const int lane = threadIdx.x & 31;
    const int wid  = threadIdx.x >> 5;
    if (lane == 0) wsum[wid] = t;
    __syncthreads();
    if (threadIdx.x == 0) {
        double s = 0.0;
#pragma unroll
        for (int i = 0; i < 8; ++i) s += (double)wsum[i];
        out[0] = (float)(s / (double)N);
    }
}

extern "C" void kernel_launch(void* const* d_in, const int* in_sizes, int n_in,
                              void* d_out, int out_size, void* d_ws, size_t ws_size,
                              hipStream_t stream) {
    const float* pred = (const float*)d_in[0];
    const float* tgt  = (const float*)d_in[1];
    float* out = (float*)d_out;

    const long long N = (long long)in_sizes[0] / 5;   // (N,5) fp32
    const int  TPB  = 256;
    const int  ROWS = 4;
    const long long rowsPerBlock = (long long)TPB * ROWS;
    const int blocks = (int)((N + rowsPerBlock - 1) / rowsPerBlock);   // 4096 for N=4M

    float* partial = (float*)d_ws;   // blocks * 4 bytes, well within ws_size

    kf_stage1<<<blocks, TPB, 0, stream>>>(pred, tgt, partial, N);
    kf_stage2<<<1, TPB, 0, stream>>>(partial, out, blocks, N);
}